// GraphConvolution_72181220376891
// MI455X (gfx1250) — compile-verified
//
#include <hip/hip_runtime.h>
#include <stdint.h>

// ---------------------------------------------------------------------------
// GraphConvolution for MI455X (gfx1250, wave32, WMMA, async-to-LDS)
//   pre0 = X @ W0 ; pre1 = X @ W1          (fp32 WMMA 16x16x4)
//   out  = scatter(adj0, pre0) + scatter(adj1, pre1)   (f32 L2 atomics)
// ---------------------------------------------------------------------------

typedef __attribute__((ext_vector_type(2))) float v2f;
typedef __attribute__((ext_vector_type(8))) float v8f;

#define N_NODES 50000
#define IN_DIM  512
#define OUT_DIM 128
#define NEDGE   600000

#define ROWS_PER_BLOCK 80     // 50000 = 625 * 80, exact
#define SUBTILES       5      // 80 / 16 row sub-tiles per wave

// LDS row stride in floats: 516*4 = 2064 bytes (16B multiple -> aligned b128
// async stores) and 516 % 64 = 4 -> the 16 rows of an A-fragment read hit
// distinct bank groups (conflict-free ds_load_b64).
#define LDS_STRIDE 516

// CDNA5 async global->LDS copy (ASYNCcnt-tracked, bypasses VGPR data path).
// Generic/flat LDS addresses carry the LDS byte offset in the low 32 bits
// (ISA 10.2 aperture rules), so truncating the pointer gives the VDST operand.
__device__ __forceinline__ void async_copy_b128(const float* gsrc, float* ldst) {
  asm volatile("global_load_async_to_lds_b128 %0, %1, off"
               :: "v"((uint32_t)(uintptr_t)ldst), "v"(gsrc)
               : "memory");
}
__device__ __forceinline__ void wait_async0() {
  asm volatile("s_wait_asynccnt 0x0" ::: "memory");
}

// ---------------------------------------------------------------------------
// GEMM: grid.x = N/80 row tiles. Block = 512 threads = 16 waves.
// Wave w owns the 16-column tile [16w, 16w+16) of the combined 256-wide
// [W0 | W1] output and computes 5 stacked 16x16 tiles (80 rows), so each
// B fragment (2 global_load_b32) feeds 5 WMMAs.
//
// f32 WMMA fragment layouts (ISA 7.12.2):
//   A 16x4 : lane L (m = L&15, kh = L>>4): VGPR0 = A[m][2kh], VGPR1 = A[m][2kh+1]
//   B 4x16 : lane L: VGPR0 = B[2kh][m],  VGPR1 = B[2kh+1][m]
//   C 16x16: VGPR r, lane L: C[r + 8*kh][m]
// ---------------------------------------------------------------------------
__global__ __launch_bounds__(512) void gemm_x_w01(
    const float* __restrict__ x,    // [N, 512]
    const float* __restrict__ w0,   // [512, 128]
    const float* __restrict__ w1,   // [512, 128]
    float* __restrict__ pre)        // [2][N][128] in workspace
{
  __shared__ float xs[ROWS_PER_BLOCK * LDS_STRIDE];   // ~165 KB of 320 KB LDS

  const int tid     = threadIdx.x;
  const int rowBase = blockIdx.x * ROWS_PER_BLOCK;

  // Stage 80 rows x 512 cols of X into LDS with async b128 copies:
  // 32 threads per row, 16 rows in flight, 5 row groups; each thread issues
  // 20 GLOBAL_LOAD_ASYNC_TO_LDS_B128 (coalesced 128B per 8 lanes).
  {
    const int r0 = tid >> 5;         // 0..15
    const int l  = tid & 31;
#pragma unroll
    for (int g = 0; g < SUBTILES; ++g) {
      const int r = r0 + 16 * g;
      const float* src = x  + (size_t)(rowBase + r) * IN_DIM;
      float*       dst = xs + r * LDS_STRIDE;
#pragma unroll
      for (int j = 0; j < 4; ++j) {
        const int c = 4 * l + 128 * j;       // covers all 512 cols
        async_copy_b128(src + c, dst + c);
      }
    }
  }
  wait_async0();      // our async writes landed in LDS
  __syncthreads();    // everyone else's too

  const int wave  = tid >> 5;        // 0..15 -> column tile
  const int lane  = tid & 31;
  const int m     = lane & 15;
  const int khalf = lane >> 4;       // 0/1 -> K sub-slot

  const int    half     = wave >> 3;               // 0: W0, 1: W1
  const int    colLocal = (wave & 7) * 16 + m;     // column within 128
  const float* W        = half ? w1 : w0;          // [512 x 128] row major

  const float* aLds = xs + m * LDS_STRIDE + 2 * khalf;              // 8B aligned
  const float* bG   = W + (size_t)(2 * khalf) * OUT_DIM + colLocal; // L2-resident

  v8f acc[SUBTILES];
#pragma unroll
  for (int s = 0; s < SUBTILES; ++s) acc[s] = (v8f){};

#pragma unroll 2
  for (int k0 = 0; k0 < IN_DIM; k0 += 4) {
    v2f b;
    b.x = bG[(size_t)(k0 + 0) * OUT_DIM];
    b.y = bG[(size_t)(k0 + 1) * OUT_DIM];
#pragma unroll
    for (int s = 0; s < SUBTILES; ++s) {
      v2f a = *(const v2f*)(aLds + (size_t)(16 * s) * LDS_STRIDE + k0);
      // 8 args: (neg_a, A, neg_b, B, c_mod, C, reuse_a, reuse_b)
      acc[s] = __builtin_amdgcn_wmma_f32_16x16x4_f32(
          false, a, false, b, (short)0, acc[s], false, false);
    }
  }

  // Store 5 C tiles: VGPR r of tile s -> row (16s + r + 8*khalf), col m.
  float* dstBase = pre + (size_t)half * ((size_t)N_NODES * OUT_DIM)
                       + (size_t)rowBase * OUT_DIM + colLocal;
#pragma unroll
  for (int s = 0; s < SUBTILES; ++s) {
    float* dst = dstBase + (size_t)(16 * s) * OUT_DIM;
#pragma unroll
    for (int r = 0; r < 8; ++r) {
      dst[(size_t)(r + 8 * khalf) * OUT_DIM] = acc[s][r];
    }
  }
}

// ---------------------------------------------------------------------------
// SpMM scatter: one wave per edge; lane handles 4 of the 128 columns.
// out[row] += val * pre[col]. pre/out are L2-resident (25.6MB each, 192MB L2)
// so gathers and f32 atomics stay on-chip.
// adjX_idx layout: flat (2,E) -> rows = idx[0..E), cols = idx[E..2E).
// ---------------------------------------------------------------------------
__global__ __launch_bounds__(256) void spmm_scatter(
    const int*   __restrict__ idx0, const float* __restrict__ val0,
    const int*   __restrict__ idx1, const float* __restrict__ val1,
    const float* __restrict__ pre,  // [2][N][128]
    float*       __restrict__ out)  // [N][128]
{
  const int gwave = (int)((blockIdx.x * blockDim.x + threadIdx.x) >> 5);
  const int lane  = threadIdx.x & 31;
  if (gwave >= 2 * NEDGE) return;   // wave-uniform (exact grid anyway)

  const int    which = (gwave >= NEDGE) ? 1 : 0;
  const int    e     = which ? (gwave - NEDGE) : gwave;
  const int*   idx   = which ? idx1 : idx0;
  const float* val   = which ? val1 : val0;
  const float* p     = pre + (size_t)which * ((size_t)N_NODES * OUT_DIM);

  const int   row = idx[e];
  const int   col = idx[NEDGE + e];
  const float v   = val[e];

  const float* src = p   + (size_t)col * OUT_DIM + lane;
  float*       dst = out + (size_t)row * OUT_DIM + lane;
#pragma unroll
  for (int j = 0; j < 4; ++j) {
    unsafeAtomicAdd(dst + 32 * j, src[32 * j] * v);  // global_atomic_add_f32
  }
}

// ---------------------------------------------------------------------------
extern "C" void kernel_launch(void* const* d_in, const int* in_sizes, int n_in,
                              void* d_out, int out_size, void* d_ws, size_t ws_size,
                              hipStream_t stream) {
  const float* x        = (const float*)d_in[0];
  const float* w0       = (const float*)d_in[1];
  const float* w1       = (const float*)d_in[2];
  const int*   adj0_idx = (const int*)  d_in[3];
  const float* adj0_val = (const float*)d_in[4];
  const int*   adj1_idx = (const int*)  d_in[5];
  const float* adj1_val = (const float*)d_in[6];

  float* out = (float*)d_out;
  float* pre = (float*)d_ws;   // needs 2*N*128*4 = 51.2 MB scratch

  // Zero accumulator (harness poisons d_out); stream memset is capturable.
  hipMemsetAsync(out, 0, (size_t)N_NODES * OUT_DIM * sizeof(float), stream);

  // Dense: 625 row tiles of 80 rows, 512 threads (16 waves) each.
  gemm_x_w01<<<N_NODES / ROWS_PER_BLOCK, 512, 0, stream>>>(x, w0, w1, pre);

  // Sparse: 2E waves, 8 waves per 256-thread block -> 150000 blocks exactly.
  spmm_scatter<<<(2 * NEDGE) / 8, 256, 0, stream>>>(
      adj0_idx, adj0_val, adj1_idx, adj1_val, pre, out);
}